// SPDRectified_13546326851717
// MI455X (gfx1250) — compile-verified
//
#include <hip/hip_runtime.h>

// SPD rectification: out = U diag(max(s,eps)) U^T per 64x64 SPD matrix.
// Matrix-function formulation:  f(A) = 0.5*(A + eps*I + |A - eps*I|),
// |M| = M*sign(M), sign via Newton-Schulz (X <- 1.5X - 0.5*X*(X*X^T)) on
// WMMA f32 16x16x4 GEMMs entirely in LDS. All B operands are symmetric,
// so B-fragments are read transposed => every fragment is one ds_load_b64.

#define EPSV 1e-4f
#define N64 64
#define S68 68               // padded LDS row stride: conflict-free b64 frags, 16B rows
#define NS_ITERS 18

typedef __attribute__((ext_vector_type(2))) float v2f;
typedef __attribute__((ext_vector_type(8))) float v8f;

// Fragment load (both A and transposed-B have the same form):
//   lane l = kh*16 + lm, vgpr v -> M[rb*16 + lm][k + 2*kh + v]  (8B aligned, conflict-free)
__device__ __forceinline__ v2f frag(const float* M, int rb, int k, int lm, int kh) {
    return *(const v2f*)&M[(rb * 16 + lm) * S68 + k + 2 * kh];
}

// One wave computes a 2x2 block of 16x16 tiles of C = Am * Bm^T (Bm symmetric).
// 4 ds_load_b64 feed 4 v_wmma per k-step.
__device__ __forceinline__ void wave_gemm2x2(const float* Am, const float* Bm,
                                             v8f acc[2][2], int sti, int stj,
                                             int lm, int kh) {
#pragma unroll
    for (int di = 0; di < 2; ++di)
#pragma unroll
        for (int dj = 0; dj < 2; ++dj)
            acc[di][dj] = (v8f){};
#pragma unroll
    for (int k = 0; k < N64; k += 4) {
        v2f a0 = frag(Am, 2 * sti + 0, k, lm, kh);
        v2f a1 = frag(Am, 2 * sti + 1, k, lm, kh);
        v2f b0 = frag(Bm, 2 * stj + 0, k, lm, kh);
        v2f b1 = frag(Bm, 2 * stj + 1, k, lm, kh);
        acc[0][0] = __builtin_amdgcn_wmma_f32_16x16x4_f32(false, a0, false, b0, (short)0, acc[0][0], false, false);
        acc[0][1] = __builtin_amdgcn_wmma_f32_16x16x4_f32(false, a0, false, b1, (short)0, acc[0][1], false, false);
        acc[1][0] = __builtin_amdgcn_wmma_f32_16x16x4_f32(false, a1, false, b0, (short)0, acc[1][0], false, false);
        acc[1][1] = __builtin_amdgcn_wmma_f32_16x16x4_f32(false, a1, false, b1, (short)0, acc[1][1], false, false);
    }
}

__global__ __launch_bounds__(128)
void spd_rectify_kernel(const float* __restrict__ in, float* __restrict__ out) {
    __shared__ float sA[N64 * S68];   // original A
    __shared__ float sX[N64 * S68];   // Newton-Schulz iterate -> sign(M)
    __shared__ float sY[N64 * S68];   // X*X^T temp; reused for staged output
    __shared__ float sRed[128];
    __shared__ float sAlpha;

    const int t    = threadIdx.x;
    const int lane = t & 31;
    const int wave = t >> 5;          // 0..3
    const int lm   = lane & 15;
    const int kh   = lane >> 4;
    const int sti  = wave >> 1;       // 2x2 super-tile coords
    const int stj  = wave & 1;
    const long long base = (long long)blockIdx.x * (N64 * N64);

    // ---- load A: 1024 float4, 8 per thread, coalesced; LDS rows 16B-aligned ----
#pragma unroll
    for (int i = 0; i < 8; ++i) {
        const int idx4 = t + i * 128;
        const int row  = idx4 >> 4;
        const int col  = (idx4 & 15) * 4;
        float4 v = *(const float4*)&in[base + row * N64 + col];
        *(float4*)&sA[row * S68 + col] = v;
    }
    __syncthreads();

    // ---- Frobenius norm of M = A - eps*I (>= spectral norm) ----
    float ss = 0.f;
#pragma unroll
    for (int i = 0; i < 32; ++i) {
        const int idx = t + i * 128;
        const int row = idx >> 6, col = idx & 63;
        float m = sA[row * S68 + col] - (row == col ? EPSV : 0.f);
        ss += m * m;
    }
    sRed[t] = ss;
    __syncthreads();
    for (int s = 64; s > 0; s >>= 1) {
        if (t < s) sRed[t] += sRed[t + s];
        __syncthreads();
    }
    if (t == 0) sAlpha = sqrtf(sRed[0]) + 1e-30f;
    __syncthreads();
    const float inva = 1.0f / sAlpha;

    // ---- X0 = M / ||M||_F  (spectrum in [-1,1]) ----
#pragma unroll
    for (int i = 0; i < 32; ++i) {
        const int idx = t + i * 128;
        const int row = idx >> 6, col = idx & 63;
        float m = sA[row * S68 + col] - (row == col ? EPSV : 0.f);
        sX[row * S68 + col] = m * inva;
    }
    __syncthreads();

    // ---- Newton-Schulz sign iteration ----
#pragma unroll 1
    for (int it = 0; it < NS_ITERS; ++it) {
        v8f acc[2][2];
        wave_gemm2x2(sX, sX, acc, sti, stj, lm, kh);   // Y = X*X^T (exactly symmetric)
#pragma unroll
        for (int di = 0; di < 2; ++di)
#pragma unroll
            for (int dj = 0; dj < 2; ++dj)
#pragma unroll
                for (int r = 0; r < 8; ++r)
                    sY[((2 * sti + di) * 16 + r + 8 * kh) * S68 +
                       (2 * stj + dj) * 16 + lm] = acc[di][dj][r];
        __syncthreads();
        wave_gemm2x2(sX, sY, acc, sti, stj, lm, kh);   // Z = X*Y
        __syncthreads();                               // all reads of sX complete
#pragma unroll
        for (int di = 0; di < 2; ++di)
#pragma unroll
            for (int dj = 0; dj < 2; ++dj)
#pragma unroll
                for (int r = 0; r < 8; ++r) {
                    const int p = ((2 * sti + di) * 16 + r + 8 * kh) * S68 +
                                  (2 * stj + dj) * 16 + lm;
                    sX[p] = 1.5f * sX[p] - 0.5f * acc[di][dj][r];   // owned position
                }
        __syncthreads();
    }

    // ---- G = A*S ; result = 0.5*(A + G) + 0.5*eps*(I - S), staged into sY ----
    {
        v8f acc[2][2];
        wave_gemm2x2(sA, sX, acc, sti, stj, lm, kh);
#pragma unroll
        for (int di = 0; di < 2; ++di)
#pragma unroll
            for (int dj = 0; dj < 2; ++dj)
#pragma unroll
                for (int r = 0; r < 8; ++r) {
                    const int row = (2 * sti + di) * 16 + r + 8 * kh;
                    const int col = (2 * stj + dj) * 16 + lm;
                    const float a = sA[row * S68 + col];
                    const float s = sX[row * S68 + col];
                    sY[row * S68 + col] = 0.5f * (a + acc[di][dj][r]) +
                                          0.5f * EPSV * ((row == col ? 1.f : 0.f) - s);
                }
    }
    __syncthreads();

    // ---- coalesced float4 writeback ----
#pragma unroll
    for (int i = 0; i < 8; ++i) {
        const int idx4 = t + i * 128;
        const int row  = idx4 >> 4;
        const int col  = (idx4 & 15) * 4;
        *(float4*)&out[base + row * N64 + col] = *(const float4*)&sY[row * S68 + col];
    }
}

extern "C" void kernel_launch(void* const* d_in, const int* in_sizes, int n_in,
                              void* d_out, int out_size, void* d_ws, size_t ws_size,
                              hipStream_t stream) {
    const float* in = (const float*)d_in[0];
    float* out = (float*)d_out;
    const int B = in_sizes[0] / (N64 * N64);   // 4096 matrices
    spd_rectify_kernel<<<dim3(B), dim3(128), 0, stream>>>(in, out);
}